// Model_38113539785435
// MI455X (gfx1250) — compile-verified
//
#include <hip/hip_runtime.h>
#include <stdint.h>

// ---------------------------------------------------------------------------
// BLAKE3 single-block compression, MI455X (gfx1250).
// Serial scalar hash: no matrix structure -> WMMA inapplicable. Optimal shape
// is one wave32, lane-uniform SALU core, fully unrolled rounds, schedule
// constant-folded, state in registers, one coalesced 128B store at the end.
// ---------------------------------------------------------------------------

constexpr uint32_t KIV[8] = {
    0x6A09E667u, 0xBB67AE85u, 0x3C6EF372u, 0xA54FF53Au,
    0x510E527Fu, 0x9B05688Cu, 0x1F83D9ABu, 0x5BE0CD19u
};

// BLAKE3 message schedule, 7 rounds (indices are compile-time constants).
constexpr int SCHED[7][16] = {
    { 0, 1, 2, 3, 4, 5, 6, 7, 8, 9,10,11,12,13,14,15},
    { 2, 6, 3,10, 7, 0, 4,13, 1,11,12, 5, 9,14,15, 8},
    { 3, 4,10,12,13, 2, 7,14, 6, 5, 9, 0,11,15, 8, 1},
    {10, 7,12, 9,14, 3,13,15, 4, 0,11, 2, 5, 8, 1, 6},
    {12,13, 9,11,15,10,14, 8, 7, 2, 5, 3, 0, 1, 6, 4},
    { 9,14,11, 5, 8,12,15, 1,13, 3, 0,10, 2, 6, 4, 7},
    {11,15, 5, 0, 1, 9, 8, 6,14,10, 2,12, 3, 4, 7,13}
};

__device__ __forceinline__ uint32_t rotr32(uint32_t x, int n) {
    return (x >> n) | (x << (32 - n));
}

// G mixing function (u32 adds wrap naturally).
#define G(a, b, c, d, mx, my)          \
    do {                               \
        a = a + b + (mx);              \
        d = rotr32(d ^ a, 16);         \
        c = c + d;                     \
        b = rotr32(b ^ c, 12);         \
        a = a + b + (my);              \
        d = rotr32(d ^ a, 8);          \
        c = c + d;                     \
        b = rotr32(b ^ c, 7);          \
    } while (0)

// One full round; r is a literal -> m[...] indices are compile-time constants,
// so m[16] stays entirely in registers (no scratch, no dynamic indexing).
#define ROUND(r)                                                      \
    do {                                                              \
        G(s0, s4, s8,  s12, m[SCHED[r][0]],  m[SCHED[r][1]]);         \
        G(s1, s5, s9,  s13, m[SCHED[r][2]],  m[SCHED[r][3]]);         \
        G(s2, s6, s10, s14, m[SCHED[r][4]],  m[SCHED[r][5]]);         \
        G(s3, s7, s11, s15, m[SCHED[r][6]],  m[SCHED[r][7]]);         \
        G(s0, s5, s10, s15, m[SCHED[r][8]],  m[SCHED[r][9]]);         \
        G(s1, s6, s11, s12, m[SCHED[r][10]], m[SCHED[r][11]]);        \
        G(s2, s7, s8,  s13, m[SCHED[r][12]], m[SCHED[r][13]]);        \
        G(s3, s4, s9,  s14, m[SCHED[r][14]], m[SCHED[r][15]]);        \
    } while (0)

__global__ __launch_bounds__(32)
void blake3_block_kernel(const int* __restrict__ msg, int* __restrict__ out)
{
    // ---- Pack 64 bytes (stored as int32 each) into 16 LE u32 words. ----
    // Addresses are wave-uniform: clang scalarizes these to SMEM s_load
    // clauses and keeps the whole packing + hash core on the SALU.
    uint32_t m[16];
#pragma unroll
    for (int w = 0; w < 16; ++w) {
        uint32_t b0 = (uint32_t)msg[4 * w + 0] & 0xFFu;
        uint32_t b1 = (uint32_t)msg[4 * w + 1] & 0xFFu;
        uint32_t b2 = (uint32_t)msg[4 * w + 2] & 0xFFu;
        uint32_t b3 = (uint32_t)msg[4 * w + 3] & 0xFFu;
        m[w] = b0 | (b1 << 8) | (b2 << 16) | (b3 << 24);
    }

    // ---- Initial state (single root block: counter=0, len=64, flags=11). ----
    uint32_t s0  = KIV[0], s1  = KIV[1], s2  = KIV[2], s3  = KIV[3];
    uint32_t s4  = KIV[4], s5  = KIV[5], s6  = KIV[6], s7  = KIV[7];
    uint32_t s8  = KIV[0], s9  = KIV[1], s10 = KIV[2], s11 = KIV[3];
    uint32_t s12 = 0u,     s13 = 0u,     s14 = 64u,    s15 = 11u;

    // ---- 7 fully unrolled rounds. ----
    ROUND(0); ROUND(1); ROUND(2); ROUND(3); ROUND(4); ROUND(5); ROUND(6);

    // ---- Finalize: h[i] = state[i] ^ state[i+8]. ----
    uint32_t h0 = s0 ^ s8,  h1 = s1 ^ s9,  h2 = s2 ^ s10, h3 = s3 ^ s11;
    uint32_t h4 = s4 ^ s12, h5 = s5 ^ s13, h6 = s6 ^ s14, h7 = s7 ^ s15;

    // ---- Fan out 32 output bytes across the 32 lanes of the wave.      ----
    // Select chain over named scalars (no indexed array -> no scratch),
    // then one coalesced 128B global_store per wave.
    uint32_t tid = threadIdx.x;           // 0..31
    uint32_t wi  = tid >> 2;              // which h word
    uint32_t hv  = (wi == 0) ? h0
                 : (wi == 1) ? h1
                 : (wi == 2) ? h2
                 : (wi == 3) ? h3
                 : (wi == 4) ? h4
                 : (wi == 5) ? h5
                 : (wi == 6) ? h6
                 :             h7;
    out[tid] = (int)((hv >> ((tid & 3u) * 8u)) & 0xFFu);
}

extern "C" void kernel_launch(void* const* d_in, const int* in_sizes, int n_in,
                              void* d_out, int out_size, void* d_ws, size_t ws_size,
                              hipStream_t stream)
{
    (void)in_sizes; (void)n_in; (void)out_size; (void)d_ws; (void)ws_size;
    const int* msg = (const int*)d_in[0];   // 64 x int32 (one byte each)
    int*       out = (int*)d_out;           // 32 x int32 (one byte each)

    // Entire problem is one serial hash: one workgroup, one wave32.
    blake3_block_kernel<<<1, 32, 0, stream>>>(msg, out);
}